// YOLOLayer_12146167513407
// MI455X (gfx1250) — compile-verified
//
#include <hip/hip_runtime.h>
#include <hip/hip_bf16.h>
#include <cstdint>

// Problem geometry (compile-time constants from the reference)
#define GG       44
#define NAA      3
#define CHN      85                 // 5 + 80 classes
#define PLANE    (GG * GG)          // 1936 spatial cells per (b, anchor) plane
#define TILE     88                 // spatial cells per workgroup tile; 1936 = 22 * 88
#define NTILES   (PLANE / TILE)     // 22
#define ELEMS    (CHN * TILE)       // 7480 elements per tile
#define ELEMS4   (ELEMS / 4)        // 1870 float4's per tile
#define QROW     (TILE / 4)         // 22 quads per channel row
#define NTHREADS 256

#if defined(__AMDGCN__) && __has_builtin(__builtin_amdgcn_tensor_load_to_lds) && \
    __has_builtin(__builtin_amdgcn_tensor_store_from_lds) && \
    __has_builtin(__builtin_amdgcn_s_wait_tensorcnt)
#define USE_TDM 1
#else
#define USE_TDM 0
#endif

typedef unsigned int u32x4 __attribute__((ext_vector_type(4)));
typedef int          i32x4 __attribute__((ext_vector_type(4)));
typedef int          i32x8 __attribute__((ext_vector_type(8)));

__global__ __launch_bounds__(NTHREADS)
void yolo_tdm_kernel(const float* __restrict__ x, float* __restrict__ out)
{
    // raw:   staged input tile, [ch][t] row-major, filled by the Tensor Data Mover
    // trans: transformed + transposed tile, [t][ch], == output layout (contiguous)
    __shared__ __align__(16) float raw[ELEMS];
    __shared__ __align__(16) float trans[ELEMS];

    const int tile = blockIdx.x;     // 0..21
    const int a    = blockIdx.y;     // anchor 0..2
    const int b    = blockIdx.z;     // batch 0..63
    const int s0   = tile * TILE;    // first spatial cell of this tile
    const int tid  = threadIdx.x;

    // Tile origin in global memory: channel 0, spatial cell s0 of plane (b, a)
    const float* gsrc = x + ((size_t)(b * (NAA * CHN) + a * CHN)) * PLANE + s0;
    // Output tile base: contiguous 7480 floats
    float* gdst = out + (((size_t)b * NAA + a) * PLANE + (size_t)s0) * CHN;

#if USE_TDM
    // --- Stage 1: one TDM DMA per workgroup pulls the 85x88 strided tile into LDS ---
    if (tid < 32) {  // wave 0 issues the DMA (EXEC ignored by TDM; branch is wave-uniform)
        uint64_t gaddr = (uint64_t)(uintptr_t)gsrc;
        uint32_t laddr = (uint32_t)(uintptr_t)(__attribute__((address_space(3))) float*)&raw[0];

        // D# group 0: count=1 (valid), lds_addr, 57-bit global_addr, type=2 ("image")
        u32x4 g0;
        g0[0] = 1u;
        g0[1] = laddr;
        g0[2] = (uint32_t)gaddr;
        g0[3] = (uint32_t)((gaddr >> 32) & 0x1FFFFFFull) | (2u << 30);

        // D# group 1: data_size=4B, tensor 1936x85 (dim0 x dim1), tile 88x85,
        //             dim0_stride=1936 elements (one channel plane), no multicast/pad/iterate
        i32x8 g1;
        g1[0] = (int)(2u << 16);                       // workgroup_mask=0 | data_size=2 (4B)
        g1[1] = (int)((uint32_t)PLANE << 16);          // tensor_dim0[15:0] at bits 63:48
        g1[2] = (int)((uint32_t)CHN << 16);            // tensor_dim0 hi=0 | tensor_dim1 lo=85
        g1[3] = (int)((uint32_t)TILE << 16);           // tensor_dim1 hi=0 | tile_dim0=88
        g1[4] = (int)CHN;                              // tile_dim1=85 | tile_dim2=0
        g1[5] = (int)PLANE;                            // tensor_dim0_stride low 32 = 1936
        g1[6] = 0;                                     // stride hi | tensor_dim1_stride lo
        g1[7] = 0;

        i32x4 gz = {0, 0, 0, 0};                       // groups 2/3 unused (2D tensor)
#if __clang_major__ >= 23
        i32x8 gz8 = {0, 0, 0, 0, 0, 0, 0, 0};
        __builtin_amdgcn_tensor_load_to_lds(g0, g1, gz, gz, gz8, 0);
#else
        __builtin_amdgcn_tensor_load_to_lds(g0, g1, gz, gz, 0);
#endif
        __builtin_amdgcn_s_wait_tensorcnt(0);
        // Compiler barrier: TDM wrote LDS behind the compiler's back (laddr is an integer)
        asm volatile("" : : "r"(laddr) : "memory");
    }
#else
    // Fallback: coalesced vector-lane loads (rows of 88 dwords are contiguous)
    for (int i = tid; i < ELEMS; i += NTHREADS) {
        int ch = i / TILE;
        int t  = i - ch * TILE;
        raw[i] = gsrc[(size_t)ch * PLANE + t];
    }
#endif
    __syncthreads();

    // Anchor sizes in output units: exp(p) * (ANCHOR/STRIDE) * STRIDE = exp(p) * ANCHOR
    const float AW = (a == 0) ? 10.0f : ((a == 1) ? 16.0f : 33.0f);
    const float AH = (a == 0) ? 13.0f : ((a == 1) ? 30.0f : 23.0f);

    // --- Stage 2: transform + transpose.  Contiguous float4 LDS reads (conflict-free);
    //     ch is uniform within each quad (88 % 4 == 0); scatter stride 85 -> gcd(21,64)=1,
    //     bank-conflict-free writes. ---
    const float4* raw4 = (const float4*)raw;
    for (int j = tid; j < ELEMS4; j += NTHREADS) {
        int ch = j / QROW;
        int t0 = (j - ch * QROW) * 4;
        float4 v4 = raw4[j];
        float vv[4] = {v4.x, v4.y, v4.z, v4.w};
#pragma unroll
        for (int q = 0; q < 4; ++q) {
            float v   = vv[q];
            float e   = __expf(v);          // v_exp_f32
            float sig = e / (e + 1.0f);     // sigmoid(v), stable for |v| < 80
            float r;
            if (ch >= 4) {                  // conf + 80 class scores: plain sigmoid
                r = sig;
            } else if (ch == 0) {           // bx: (sigmoid + grid_x) * stride(8)
                int s = s0 + t0 + q;
                r = (sig + (float)(s % GG)) * 8.0f;
            } else if (ch == 1) {           // by: (sigmoid + grid_y) * stride(8)
                int s = s0 + t0 + q;
                r = (sig + (float)(s / GG)) * 8.0f;
            } else if (ch == 2) {           // bw * stride = exp(p) * anchor_w
                r = e * AW;
            } else {                        // bh * stride = exp(p) * anchor_h
                r = e * AH;
            }
            trans[(t0 + q) * CHN + ch] = r;
        }
    }
    __syncthreads();   // all trans[] writes complete & visible before the DMA reads LDS

#if USE_TDM
    // --- Stage 3: one TDM DMA per workgroup streams the contiguous 29,920 B tile out.
    //     1-row tile: tensor_dim0 = tile_dim0 = 7480 dwords, tile_dim1 = 1. ---
    if (tid < 32) {
        uint64_t gaddr = (uint64_t)(uintptr_t)gdst;
        uint32_t laddr = (uint32_t)(uintptr_t)(__attribute__((address_space(3))) float*)&trans[0];

        u32x4 g0;
        g0[0] = 1u;                                    // count=1
        g0[1] = laddr;
        g0[2] = (uint32_t)gaddr;
        g0[3] = (uint32_t)((gaddr >> 32) & 0x1FFFFFFull) | (2u << 30);   // type=2

        i32x8 g1;
        g1[0] = (int)(2u << 16);                       // data_size = 4 bytes
        g1[1] = (int)(((uint32_t)ELEMS & 0xFFFFu) << 16);  // tensor_dim0 lo16 = 7480
        g1[2] = (int)(1u << 16);                       // tensor_dim0 hi=0 | tensor_dim1 = 1
        g1[3] = (int)((uint32_t)ELEMS << 16);          // tensor_dim1 hi=0 | tile_dim0 = 7480
        g1[4] = 1;                                     // tile_dim1 = 1 | tile_dim2 = 0
        g1[5] = (int)ELEMS;                            // tensor_dim0_stride = 7480 (unused, 1 row)
        g1[6] = 0;
        g1[7] = 0;

        i32x4 gz = {0, 0, 0, 0};
        // Force all prior LDS stores to be considered live before the opaque DMA reads them
        asm volatile("" : : "r"(laddr) : "memory");
#if __clang_major__ >= 23
        i32x8 gz8 = {0, 0, 0, 0, 0, 0, 0, 0};
        __builtin_amdgcn_tensor_store_from_lds(g0, g1, gz, gz, gz8, 0);
#else
        __builtin_amdgcn_tensor_store_from_lds(g0, g1, gz, gz, 0);
#endif
        __builtin_amdgcn_s_wait_tensorcnt(0);   // DMA complete before wave 0 retires
    }
#else
    // Fallback: coalesced 128-bit stores of the contiguous output tile
    const float4* tr4 = (const float4*)trans;
    float4*       o4  = (float4*)gdst;
    for (int j = tid; j < ELEMS4; j += NTHREADS) {
        o4[j] = tr4[j];
    }
#endif
}

extern "C" void kernel_launch(void* const* d_in, const int* in_sizes, int n_in,
                              void* d_out, int out_size, void* d_ws, size_t ws_size,
                              hipStream_t stream)
{
    const float* x   = (const float*)d_in[0];   // (64, 255, 44, 44) f32
    float*       out = (float*)d_out;           // (64, 5808, 85) f32
    dim3 grid(NTILES, NAA, 64);                 // 22 x 3 x 64 = 4224 workgroups
    yolo_tdm_kernel<<<grid, NTHREADS, 0, stream>>>(x, out);
}